// ZenithHolographicVisualEncoder_3195455668787
// MI455X (gfx1250) — compile-verified
//
#include <hip/hip_runtime.h>
#include <hip/hip_bf16.h>

// ---------------------------------------------------------------------------
// ZenithHolographicVisualEncoder for gfx1250 (MI455X), wave32 + WMMA f32.
// FFT2 computed as Z = F * X * F (F = symmetric 128x128 DFT matrix), all
// dense linear algebra on V_WMMA_F32_16X16X4_F32 with 32x32 register
// blocking; stage-1 image panel staged to LDS via global_load_async_to_lds.
// ---------------------------------------------------------------------------

typedef __attribute__((ext_vector_type(2))) float v2f;
typedef __attribute__((ext_vector_type(8))) float v8f;

#define WMMA4(a, b, c) \
  __builtin_amdgcn_wmma_f32_16x16x4_f32(false, (a), false, (b), (short)0, (c), false, false)

// ---- problem constants ----
#define BB   128
#define CC   2
#define HH   128
#define WW   128
#define DD   32768      // C*H*W
#define TT   64         // num pattern types
#define NPLANE 256      // B*C
#define PLANE  16384    // H*W

// ---- workspace layout (float elements) ----
static const size_t OFF_FR   = 0;                       // 128*128
static const size_t OFF_FI   = OFF_FR  + 16384;         // 128*128
static const size_t OFF_YR   = OFF_FI  + 16384;         // 256*16384
static const size_t OFF_YI   = OFF_YR  + 4194304;
static const size_t OFF_MAG  = OFF_YI  + 4194304;       // 256*16384 == [B, D]
static const size_t OFF_NRM  = OFF_MAG + 4194304;       // 128
static const size_t OFF_RNRM = OFF_NRM + 128;           // 64
static const size_t OFF_FM   = OFF_RNRM + 64;           // 128*128
static const size_t OFF_HID  = OFF_FM  + 16384;         // 128*512
static const size_t OFF_SIMS = OFF_HID + 65536;         // 128*64
static const size_t OFF_ACTS = OFF_SIMS + 8192;         // 128*64
static const size_t OFF_P1   = OFF_ACTS + 8192;         // 128*32*64*64
static const size_t OFF_S    = OFF_P1  + 16777216;      // 128*64
static const size_t OFF_Z    = OFF_S   + 8192;          // 128*32768
static const size_t OFF_MU   = OFF_Z   + 4194304;       // 128
static const size_t OFF_RSTD = OFF_MU  + 128;           // 128
static const size_t OFF_FN   = OFF_RSTD + 128;          // 128*128  (-Fi)

// ---------------------------------------------------------------------------
// Fragment loaders. A 16x4 f32: lane m = lane&15, kb = (lane>>4)*2.
// B 4x16 f32: lane n = lane&15, same kb. C/D: lane n, VGPR v -> M = v+8*hi.
// ---------------------------------------------------------------------------
__device__ __forceinline__ v2f ldA(const float* __restrict__ A, int lda,
                                   int k, int m, int kb) {
  v2f a;
  a.x = A[m * lda + k + kb];
  a.y = A[m * lda + k + kb + 1];
  return a;
}
__device__ __forceinline__ v2f ldB(const float* __restrict__ B, int ldb,
                                   int k, int n, int kb) {
  v2f b;
  b.x = B[(k + kb) * ldb + n];
  b.y = B[(k + kb + 1) * ldb + n];
  return b;
}
// B transposed: Bt is N x K row-major, B[k][n] = Bt[n*ldb + k].
__device__ __forceinline__ v2f ldBt(const float* __restrict__ Bt, int ldb,
                                    int k, int n, int kb) {
  v2f b;
  b.x = Bt[n * ldb + k + kb];
  b.y = Bt[n * ldb + k + kb + 1];
  return b;
}

__device__ __forceinline__ void store_tile(float* __restrict__ C, int ldc, v8f c) {
  const int lane = threadIdx.x & 31;
  const int n  = lane & 15;
  const int mo = (lane >> 4) << 3;
#pragma unroll
  for (int v = 0; v < 8; ++v) C[(mo + v) * ldc + n] = c[v];
}

// Epilogue over one 16x16 C tile: EXPR sees (row, col, val).
#define TILE_EPILOGUE(ACC, TM, TN, EXPR)                                    \
  {                                                                         \
    const int lane_ = threadIdx.x & 31;                                     \
    const int n_ = lane_ & 15;                                              \
    const int mo_ = (lane_ >> 4) << 3;                                      \
    _Pragma("unroll") for (int v = 0; v < 8; ++v) {                         \
      const int row = (TM) + mo_ + v;                                       \
      const int col = (TN) + n_;                                            \
      const float val = (ACC)[v];                                           \
      EXPR;                                                                 \
    }                                                                       \
  }

__device__ __forceinline__ float block_reduce_sum(float v, float* red) {
  const int tid = threadIdx.x;
  __syncthreads();               // safe reuse of `red` across calls
  red[tid] = v;
  __syncthreads();
  for (int s = 128; s > 0; s >>= 1) {
    if (tid < s) red[tid] += red[tid + s];
    __syncthreads();
  }
  return red[0];
}

// ---------------------------------------------------------------------------
// K0: DFT twiddle matrices. F symmetric; Fn = -Fi so Zr accumulates add-only.
// ---------------------------------------------------------------------------
__global__ void make_twiddles(float* __restrict__ Fr, float* __restrict__ Fi,
                              float* __restrict__ Fn) {
  int idx = blockIdx.x * blockDim.x + threadIdx.x;
  if (idx >= 16384) return;
  int j = idx >> 7, k = idx & 127;
  float ang = -6.283185307179586f * (float)((j * k) & 127) * (1.0f / 128.0f);
  float c = cosf(ang), s = sinf(ang);
  Fr[idx] = c;
  Fi[idx] = s;
  Fn[idx] = -s;
}

// ---------------------------------------------------------------------------
// K1: Y = F @ X per plane. Image panel staged to LDS with async-to-LDS copy
// (ASYNCcnt path), then Yr/Yi computed together: 8 WMMAs per 6 fragments.
// The LDS destination operand is derived from the real pointer (ptrtoint
// captures sP, so the compiler must assume the asm writes it -- the "m"
// constraint route crashes the backend, and a raw integer offset lets LDS
// loads fold to undef).
// ---------------------------------------------------------------------------
__global__ __launch_bounds__(256) void dft_stage1(const float* __restrict__ img,
                                                  const float* __restrict__ Fr,
                                                  const float* __restrict__ Fi,
                                                  float* __restrict__ Yr,
                                                  float* __restrict__ Yi) {
  __shared__ float sP[PLANE];                       // 64 KB image plane
  const int plane = blockIdx.x;
  const float* P = img + (size_t)plane * PLANE;

  // Async copy 64KB plane -> LDS: 4096 x b128, 16 per thread, ASYNCcnt-tracked.
  {
    const unsigned tid = threadIdx.x;
#pragma unroll
    for (int i = 0; i < 16; ++i) {
      const unsigned slot = tid + i * 256u;          // float4 slot
      float* ldst = sP + slot * 4u;
      // Low 32 bits of a generic LDS address are the LDS byte offset.
      unsigned lds = (unsigned)(uintptr_t)ldst;
      unsigned long long ga = (unsigned long long)(uintptr_t)(P + slot * 4u);
      asm volatile("global_load_async_to_lds_b128 %0, %1, off"
                   :: "v"(lds), "v"(ga) : "memory");
    }
    asm volatile("s_wait_asynccnt 0x0" ::: "memory");
    // Zero-instruction escape: keeps sP visibly written by the async engine.
    asm volatile("" :: "v"((void*)sP) : "memory");
  }
  __syncthreads();

  float* yr = Yr + (size_t)plane * PLANE;
  float* yi = Yi + (size_t)plane * PLANE;
  const int wave = threadIdx.x >> 5;
  const int lane = threadIdx.x & 31;
  const int m = lane & 15, n = lane & 15, kb = (lane >> 4) << 1;

  for (int bk = wave; bk < 16; bk += 8) {            // 4x4 grid of 32x32 blocks
    const int bm = (bk >> 2) << 5;
    const int bn = (bk & 3) << 5;
    const float* fr0p = Fr + bm * 128;
    const float* fr1p = Fr + (bm + 16) * 128;
    const float* fi0p = Fi + bm * 128;
    const float* fi1p = Fi + (bm + 16) * 128;
    v8f r00 = {}, r01 = {}, r10 = {}, r11 = {};
    v8f i00 = {}, i01 = {}, i10 = {}, i11 = {};
    for (int k = 0; k < 128; k += 4) {
      v2f ar0 = ldA(fr0p, 128, k, m, kb), ar1 = ldA(fr1p, 128, k, m, kb);
      v2f ai0 = ldA(fi0p, 128, k, m, kb), ai1 = ldA(fi1p, 128, k, m, kb);
      v2f b0 = ldB(sP + bn, 128, k, n, kb);
      v2f b1 = ldB(sP + bn + 16, 128, k, n, kb);
      r00 = WMMA4(ar0, b0, r00); r01 = WMMA4(ar0, b1, r01);
      r10 = WMMA4(ar1, b0, r10); r11 = WMMA4(ar1, b1, r11);
      i00 = WMMA4(ai0, b0, i00); i01 = WMMA4(ai0, b1, i01);
      i10 = WMMA4(ai1, b0, i10); i11 = WMMA4(ai1, b1, i11);
    }
    store_tile(yr + bm * 128 + bn, 128, r00);
    store_tile(yr + bm * 128 + bn + 16, 128, r01);
    store_tile(yr + (bm + 16) * 128 + bn, 128, r10);
    store_tile(yr + (bm + 16) * 128 + bn + 16, 128, r11);
    store_tile(yi + bm * 128 + bn, 128, i00);
    store_tile(yi + bm * 128 + bn + 16, 128, i01);
    store_tile(yi + (bm + 16) * 128 + bn, 128, i10);
    store_tile(yi + (bm + 16) * 128 + bn + 16, 128, i11);
  }
}

// ---------------------------------------------------------------------------
// K2: Z = Y @ F, mag = |Z|.  Zr = Yr@Fr + Yi@Fn ; Zi = Yr@Fi + Yi@Fr.
// 16 WMMAs per 10 fragments per k-step (32x32 block, Zr+Zi fused).
// ---------------------------------------------------------------------------
__global__ __launch_bounds__(256) void dft_stage2(const float* __restrict__ Yr,
                                                  const float* __restrict__ Yi,
                                                  const float* __restrict__ Fr,
                                                  const float* __restrict__ Fi,
                                                  const float* __restrict__ Fn,
                                                  float* __restrict__ mag) {
  const int plane = blockIdx.x;
  const float* yr = Yr + (size_t)plane * PLANE;
  const float* yi = Yi + (size_t)plane * PLANE;
  float* mg = mag + (size_t)plane * PLANE;
  const int wave = threadIdx.x >> 5;
  const int lane = threadIdx.x & 31;
  const int m = lane & 15, n = lane & 15, kb = (lane >> 4) << 1;

  for (int bk = wave; bk < 16; bk += 8) {
    const int bm = (bk >> 2) << 5;
    const int bn = (bk & 3) << 5;
    const float* yr0p = yr + bm * 128;
    const float* yr1p = yr + (bm + 16) * 128;
    const float* yi0p = yi + bm * 128;
    const float* yi1p = yi + (bm + 16) * 128;
    v8f zr00 = {}, zr01 = {}, zr10 = {}, zr11 = {};
    v8f zi00 = {}, zi01 = {}, zi10 = {}, zi11 = {};
    for (int k = 0; k < 128; k += 4) {
      v2f a_r0 = ldA(yr0p, 128, k, m, kb), a_r1 = ldA(yr1p, 128, k, m, kb);
      v2f a_i0 = ldA(yi0p, 128, k, m, kb), a_i1 = ldA(yi1p, 128, k, m, kb);
      v2f fr0 = ldB(Fr + bn, 128, k, n, kb), fr1 = ldB(Fr + bn + 16, 128, k, n, kb);
      v2f fi0 = ldB(Fi + bn, 128, k, n, kb), fi1 = ldB(Fi + bn + 16, 128, k, n, kb);
      v2f fn0 = ldB(Fn + bn, 128, k, n, kb), fn1 = ldB(Fn + bn + 16, 128, k, n, kb);
      zr00 = WMMA4(a_r0, fr0, zr00); zr00 = WMMA4(a_i0, fn0, zr00);
      zr01 = WMMA4(a_r0, fr1, zr01); zr01 = WMMA4(a_i0, fn1, zr01);
      zr10 = WMMA4(a_r1, fr0, zr10); zr10 = WMMA4(a_i1, fn0, zr10);
      zr11 = WMMA4(a_r1, fr1, zr11); zr11 = WMMA4(a_i1, fn1, zr11);
      zi00 = WMMA4(a_r0, fi0, zi00); zi00 = WMMA4(a_i0, fr0, zi00);
      zi01 = WMMA4(a_r0, fi1, zi01); zi01 = WMMA4(a_i0, fr1, zi01);
      zi10 = WMMA4(a_r1, fi0, zi10); zi10 = WMMA4(a_i1, fr0, zi10);
      zi11 = WMMA4(a_r1, fi1, zi11); zi11 = WMMA4(a_i1, fr1, zi11);
    }
    TILE_EPILOGUE(zr00, bm, bn,
                  mg[row * 128 + col] = sqrtf(val * val + zi00[v] * zi00[v]));
    TILE_EPILOGUE(zr01, bm, bn + 16,
                  mg[row * 128 + col] = sqrtf(val * val + zi01[v] * zi01[v]));
    TILE_EPILOGUE(zr10, bm + 16, bn,
                  mg[row * 128 + col] = sqrtf(val * val + zi10[v] * zi10[v]));
    TILE_EPILOGUE(zr11, bm + 16, bn + 16,
                  mg[row * 128 + col] = sqrtf(val * val + zi11[v] * zi11[v]));
  }
}

// ---------------------------------------------------------------------------
// K3: per-batch L2 norm of mag row + 8x8 adaptive avg pool (fm, [B, 128]).
// ---------------------------------------------------------------------------
__global__ __launch_bounds__(256) void mag_stats(const float* __restrict__ mag,
                                                 float* __restrict__ norms,
                                                 float* __restrict__ fm) {
  const int b = blockIdx.x;
  const float* m = mag + (size_t)b * DD;
  __shared__ float red[256];
  float ss = 0.f;
  for (int i = threadIdx.x; i < DD; i += 256) {
    float v = m[i];
    ss += v * v;
  }
  float tot = block_reduce_sum(ss, red);
  if (threadIdx.x == 0) norms[b] = sqrtf(tot);
  if (threadIdx.x < 128) {
    const int c = threadIdx.x >> 6;
    const int cell = threadIdx.x & 63;
    const int i = cell >> 3, j = cell & 7;
    const float* base = m + c * PLANE + (i * 16) * 128 + j * 16;
    float s = 0.f;
    for (int h = 0; h < 16; ++h)
      for (int w = 0; w < 16; ++w) s += base[h * 128 + w];
    fm[b * 128 + threadIdx.x] = s * (1.0f / 256.0f);
  }
}

__global__ __launch_bounds__(256) void recept_norms(const float* __restrict__ R,
                                                    float* __restrict__ rn) {
  const int t = blockIdx.x;
  const float* r = R + (size_t)t * DD;
  __shared__ float red[256];
  float ss = 0.f;
  for (int i = threadIdx.x; i < DD; i += 256) {
    float v = r[i];
    ss += v * v;
  }
  float tot = block_reduce_sum(ss, red);
  if (threadIdx.x == 0) rn[t] = sqrtf(tot);
}

// ---------------------------------------------------------------------------
// K4: sims = normalize(mag_flat) @ normalize(receptors)^T  [128x64], K=32768.
// 32x32 blocks, NT layout (both operands K-contiguous -> b64 loads).
// ---------------------------------------------------------------------------
__global__ __launch_bounds__(256) void sims_gemm(const float* __restrict__ mag,
                                                 const float* __restrict__ R,
                                                 const float* __restrict__ norms,
                                                 const float* __restrict__ rnorms,
                                                 float* __restrict__ sims) {
  const int wave = threadIdx.x >> 5;                 // 8 waves = 8 blocks (4x2)
  const int bm = (wave >> 1) << 5;
  const int bn = (wave & 1) << 5;
  const int lane = threadIdx.x & 31;
  const int m = lane & 15, n = lane & 15, kb = (lane >> 4) << 1;
  const float* A0 = mag + (size_t)bm * DD;
  const float* A1 = mag + (size_t)(bm + 16) * DD;
  const float* B0 = R + (size_t)bn * DD;
  const float* B1 = R + (size_t)(bn + 16) * DD;
  v8f c00 = {}, c01 = {}, c10 = {}, c11 = {};
  for (int k = 0; k < DD; k += 4) {
    v2f a0 = ldA(A0, DD, k, m, kb), a1 = ldA(A1, DD, k, m, kb);
    v2f b0 = ldBt(B0, DD, k, n, kb), b1 = ldBt(B1, DD, k, n, kb);
    c00 = WMMA4(a0, b0, c00); c01 = WMMA4(a0, b1, c01);
    c10 = WMMA4(a1, b0, c10); c11 = WMMA4(a1, b1, c11);
  }
#define SIMS_STORE \
  sims[row * TT + col] = val / ((norms[row] + 1e-12f) * (rnorms[col] + 1e-12f))
  TILE_EPILOGUE(c00, bm, bn, SIMS_STORE);
  TILE_EPILOGUE(c01, bm, bn + 16, SIMS_STORE);
  TILE_EPILOGUE(c10, bm + 16, bn, SIMS_STORE);
  TILE_EPILOGUE(c11, bm + 16, bn + 16, SIMS_STORE);
#undef SIMS_STORE
}

// ---------------------------------------------------------------------------
// K5: zenith sparse attention (d=1) + acts = softmax(10*att). 1 block/batch.
// ---------------------------------------------------------------------------
__global__ __launch_bounds__(64) void attention_acts(const float* __restrict__ sims,
                                                     const float* __restrict__ wq,
                                                     const float* __restrict__ wk,
                                                     const float* __restrict__ wv,
                                                     const float* __restrict__ wo,
                                                     float* __restrict__ acts_out,
                                                     float* __restrict__ acts_ws) {
  const int b = blockIdx.x;
  const int i = threadIdx.x;          // query index 0..63
  __shared__ float srow[64];
  __shared__ float att[64];
  srow[i] = sims[b * TT + i];
  __syncthreads();
  const float q = srow[i] * wq[0];
  const float kw = wk[0], vw = wv[0];
  float sc[64];
  for (int j = 0; j < 64; ++j) sc[j] = q * (srow[j] * kw);   // d=1, sqrt(d)=1
  // 16-th largest (counting duplicates) via repeated extraction
  float tmp[64];
  for (int j = 0; j < 64; ++j) tmp[j] = sc[j];
  float kth = -__builtin_inff();
  for (int t = 0; t < 16; ++t) {
    float mx = -__builtin_inff();
    int mi = 0;
    for (int j = 0; j < 64; ++j)
      if (tmp[j] > mx) { mx = tmp[j]; mi = j; }
    kth = mx;
    tmp[mi] = -__builtin_inff();
  }
  // masked softmax over kept scores
  float mx = -__builtin_inff();
  for (int j = 0; j < 64; ++j)
    if (sc[j] >= kth && sc[j] > mx) mx = sc[j];
  float w[64];
  float sum = 0.f;
  for (int j = 0; j < 64; ++j) {
    float e = (sc[j] >= kth) ? expf(sc[j] - mx) : 0.f;
    w[j] = e;
    sum += e;
  }
  const float inv = 1.f / sum;
  float wmax = 0.f;
  for (int j = 0; j < 64; ++j) {
    w[j] *= inv;
    wmax = fmaxf(wmax, w[j]);
  }
  const float thr = 0.2f * wmax;       // (1 - 0.8) * wmax
  float s2 = 0.f;
  for (int j = 0; j < 64; ++j) {
    if (w[j] < thr) w[j] = 0.f;
    s2 += w[j];
  }
  float o = 0.f;
  for (int j = 0; j < 64; ++j) o += w[j] * (srow[j] * vw);
  o = o / (s2 + 1e-9f) * wo[0];
  att[i] = o;
  __syncthreads();
  float m10 = -__builtin_inff();
  for (int j = 0; j < 64; ++j) m10 = fmaxf(m10, att[j]);
  float se = 0.f;
  for (int j = 0; j < 64; ++j) se += expf(10.f * (att[j] - m10));
  const float a = expf(10.f * (att[i] - m10)) / se;
  acts_out[b * TT + i] = a;
  acts_ws[b * TT + i] = a;
}

// ---------------------------------------------------------------------------
// K6: conv1(2->32, 3x3 SAME) + relu + maxpool2 fused. out: [B,32,64,64]
// ---------------------------------------------------------------------------
__global__ __launch_bounds__(256) void conv1_pool(const float* __restrict__ img,
                                                  const float* __restrict__ W,
                                                  const float* __restrict__ bias,
                                                  float* __restrict__ pool1) {
  const int b = blockIdx.x >> 5;
  const int oc = blockIdx.x & 31;
  const float* x0 = img + (size_t)b * CC * PLANE;
  const float* wgt = W + oc * CC * 9;
  const float bb = bias[oc];
  float* out = pool1 + (((size_t)b * 32 + oc) << 12);   // 64*64
  for (int p = threadIdx.x; p < 4096; p += 256) {
    const int oh = p >> 6, ow = p & 63;
    float mx = -__builtin_inff();
#pragma unroll
    for (int dy = 0; dy < 2; ++dy)
#pragma unroll
      for (int dx = 0; dx < 2; ++dx) {
        const int h = oh * 2 + dy, w = ow * 2 + dx;
        float acc = bb;
        for (int ic = 0; ic < 2; ++ic) {
          const float* xc = x0 + ic * PLANE;
          const float* wc = wgt + ic * 9;
#pragma unroll
          for (int kh = 0; kh < 3; ++kh) {
            const int ih = h + kh - 1;
            if (ih < 0 || ih >= 128) continue;
#pragma unroll
            for (int kw = 0; kw < 3; ++kw) {
              const int iw = w + kw - 1;
              if (iw < 0 || iw >= 128) continue;
              acc += xc[ih * 128 + iw] * wc[kh * 3 + kw];
            }
          }
        }
        mx = fmaxf(mx, fmaxf(acc, 0.f));
      }
    out[p] = mx;
  }
}

// ---------------------------------------------------------------------------
// K7: conv2(32->64, 3x3 SAME) + relu + global avg pool. out: s [B, 64]
// ---------------------------------------------------------------------------
__global__ __launch_bounds__(256) void conv2_gap(const float* __restrict__ pool1,
                                                 const float* __restrict__ W,
                                                 const float* __restrict__ bias,
                                                 float* __restrict__ s_feat) {
  const int b = blockIdx.x >> 6;
  const int oc = blockIdx.x & 63;
  const float* x0 = pool1 + (size_t)b * 32 * 4096;
  const float* wgt = W + oc * 32 * 9;
  const float bb = bias[oc];
  __shared__ float red[256];
  float part = 0.f;
  for (int p = threadIdx.x; p < 4096; p += 256) {
    const int oh = p >> 6, ow = p & 63;
    float acc = bb;
    for (int ic = 0; ic < 32; ++ic) {
      const float* xc = x0 + ic * 4096;
      const float* wc = wgt + ic * 9;
#pragma unroll
      for (int kh = 0; kh < 3; ++kh) {
        const int ih = oh + kh - 1;
        if (ih < 0 || ih >= 64) continue;
#pragma unroll
        for (int kw = 0; kw < 3; ++kw) {
          const int iw = ow + kw - 1;
          if (iw < 0 || iw >= 64) continue;
          acc += xc[ih * 64 + iw] * wc[kh * 3 + kw];
        }
      }
    }
    part += fmaxf(acc, 0.f);
  }
  float tot = block_reduce_sum(part, red);
  if (threadIdx.x == 0) s_feat[b * 64 + oc] = tot * (1.0f / 4096.0f);
}

// ---------------------------------------------------------------------------
// K8: hidden = relu(fm @ fp_w1 + b1)  [128 x 512], K=128, 32x32 blocks.
// ---------------------------------------------------------------------------
__global__ __launch_bounds__(256) void fp1_gemm(const float* __restrict__ fm,
                                                const float* __restrict__ W1,
                                                const float* __restrict__ b1,
                                                float* __restrict__ hidden) {
  const int blk = blockIdx.x * 8 + (threadIdx.x >> 5);  // 0..63 (4x16)
  const int bm = (blk >> 4) << 5;
  const int bn = (blk & 15) << 5;
  const int lane = threadIdx.x & 31;
  const int m = lane & 15, n = lane & 15, kb = (lane >> 4) << 1;
  v8f c00 = {}, c01 = {}, c10 = {}, c11 = {};
  for (int k = 0; k < 128; k += 4) {
    v2f a0 = ldA(fm + bm * 128, 128, k, m, kb);
    v2f a1 = ldA(fm + (bm + 16) * 128, 128, k, m, kb);
    v2f b0 = ldB(W1 + bn, 512, k, n, kb);
    v2f b1v = ldB(W1 + bn + 16, 512, k, n, kb);
    c00 = WMMA4(a0, b0, c00); c01 = WMMA4(a0, b1v, c01);
    c10 = WMMA4(a1, b0, c10); c11 = WMMA4(a1, b1v, c11);
  }
#define FP1_STORE hidden[row * 512 + col] = fmaxf(val + b1[col], 0.f)
  TILE_EPILOGUE(c00, bm, bn, FP1_STORE);
  TILE_EPILOGUE(c01, bm, bn + 16, FP1_STORE);
  TILE_EPILOGUE(c10, bm + 16, bn, FP1_STORE);
  TILE_EPILOGUE(c11, bm + 16, bn + 16, FP1_STORE);
#undef FP1_STORE
}

// ---------------------------------------------------------------------------
// K9: z = hidden @ fp_w2 + b2   [128 x 32768], K=512, 32x32 blocks.
// ---------------------------------------------------------------------------
__global__ __launch_bounds__(256) void fp2_gemm(const float* __restrict__ hidden,
                                                const float* __restrict__ W2,
                                                const float* __restrict__ b2,
                                                float* __restrict__ z) {
  const int blk = blockIdx.x * 8 + (threadIdx.x >> 5);  // 0..4095 (4x1024)
  const int bm = (blk >> 10) << 5;
  const int bn = (blk & 1023) << 5;
  const int lane = threadIdx.x & 31;
  const int m = lane & 15, n = lane & 15, kb = (lane >> 4) << 1;
  v8f c00 = {}, c01 = {}, c10 = {}, c11 = {};
  for (int k = 0; k < 512; k += 4) {
    v2f a0 = ldA(hidden + bm * 512, 512, k, m, kb);
    v2f a1 = ldA(hidden + (bm + 16) * 512, 512, k, m, kb);
    v2f b0 = ldB(W2 + bn, DD, k, n, kb);
    v2f b1v = ldB(W2 + bn + 16, DD, k, n, kb);
    c00 = WMMA4(a0, b0, c00); c01 = WMMA4(a0, b1v, c01);
    c10 = WMMA4(a1, b0, c10); c11 = WMMA4(a1, b1v, c11);
  }
#define FP2_STORE z[(size_t)row * DD + col] = val + b2[col]
  TILE_EPILOGUE(c00, bm, bn, FP2_STORE);
  TILE_EPILOGUE(c01, bm, bn + 16, FP2_STORE);
  TILE_EPILOGUE(c10, bm + 16, bn, FP2_STORE);
  TILE_EPILOGUE(c11, bm + 16, bn + 16, FP2_STORE);
#undef FP2_STORE
}

// ---------------------------------------------------------------------------
// K10: LayerNorm row stats of z (mean / rsqrt(var+1e-5)) per batch row.
// ---------------------------------------------------------------------------
__global__ __launch_bounds__(256) void ln_stats(const float* __restrict__ z,
                                                float* __restrict__ mu,
                                                float* __restrict__ rstd) {
  const int b = blockIdx.x;
  const float* zr = z + (size_t)b * DD;
  __shared__ float red[256];
  float s = 0.f, sq = 0.f;
  for (int i = threadIdx.x; i < DD; i += 256) {
    float v = zr[i];
    s += v;
    sq += v * v;
  }
  float ts = block_reduce_sum(s, red);
  float tq = block_reduce_sum(sq, red);
  if (threadIdx.x == 0) {
    float m = ts * (1.0f / DD);
    float var = tq * (1.0f / DD) - m * m;
    mu[b] = m;
    rstd[b] = rsqrtf(var + 1e-5f);
  }
}

// ---------------------------------------------------------------------------
// K11: holo = LN(z)*g+b + acts@receptors + s@sp_w + sp_b   [128 x 32768]
// Two K=64 WMMA accumulations fused with the layernorm epilogue, 32x32 blocks.
// ---------------------------------------------------------------------------
__global__ __launch_bounds__(256) void holo_gemm(const float* __restrict__ z,
                                                 const float* __restrict__ mu,
                                                 const float* __restrict__ rstd,
                                                 const float* __restrict__ ln_g,
                                                 const float* __restrict__ ln_b,
                                                 const float* __restrict__ acts,
                                                 const float* __restrict__ R,
                                                 const float* __restrict__ s_feat,
                                                 const float* __restrict__ sp_w,
                                                 const float* __restrict__ sp_b,
                                                 float* __restrict__ holo) {
  const int blk = blockIdx.x * 8 + (threadIdx.x >> 5);  // 0..4095 (4x1024)
  const int bm = (blk >> 10) << 5;
  const int bn = (blk & 1023) << 5;
  const int lane = threadIdx.x & 31;
  const int m = lane & 15, n = lane & 15, kb = (lane >> 4) << 1;
  v8f c00 = {}, c01 = {}, c10 = {}, c11 = {};
  for (int k = 0; k < TT; k += 4) {                 // acts @ receptors
    v2f a0 = ldA(acts + bm * TT, TT, k, m, kb);
    v2f a1 = ldA(acts + (bm + 16) * TT, TT, k, m, kb);
    v2f b0 = ldB(R + bn, DD, k, n, kb);
    v2f b1v = ldB(R + bn + 16, DD, k, n, kb);
    c00 = WMMA4(a0, b0, c00); c01 = WMMA4(a0, b1v, c01);
    c10 = WMMA4(a1, b0, c10); c11 = WMMA4(a1, b1v, c11);
  }
  for (int k = 0; k < TT; k += 4) {                 // s @ sp_w
    v2f a0 = ldA(s_feat + bm * TT, TT, k, m, kb);
    v2f a1 = ldA(s_feat + (bm + 16) * TT, TT, k, m, kb);
    v2f b0 = ldB(sp_w + bn, DD, k, n, kb);
    v2f b1v = ldB(sp_w + bn + 16, DD, k, n, kb);
    c00 = WMMA4(a0, b0, c00); c01 = WMMA4(a0, b1v, c01);
    c10 = WMMA4(a1, b0, c10); c11 = WMMA4(a1, b1v, c11);
  }
#define HOLO_STORE                                                          \
  {                                                                         \
    const float zz = z[(size_t)row * DD + col];                             \
    const float zn = (zz - mu[row]) * rstd[row] * ln_g[col] + ln_b[col];    \
    holo[(size_t)row * DD + col] = zn + val + sp_b[col];                    \
  }
  TILE_EPILOGUE(c00, bm, bn, HOLO_STORE);
  TILE_EPILOGUE(c01, bm, bn + 16, HOLO_STORE);
  TILE_EPILOGUE(c10, bm + 16, bn, HOLO_STORE);
  TILE_EPILOGUE(c11, bm + 16, bn + 16, HOLO_STORE);
#undef HOLO_STORE
}

// ---------------------------------------------------------------------------
extern "C" void kernel_launch(void* const* d_in, const int* in_sizes, int n_in,
                              void* d_out, int out_size, void* d_ws, size_t ws_size,
                              hipStream_t stream) {
  const float* image     = (const float*)d_in[0];
  const float* receptors = (const float*)d_in[1];
  const float* fp_w1     = (const float*)d_in[2];
  const float* fp_b1     = (const float*)d_in[3];
  const float* fp_w2     = (const float*)d_in[4];
  const float* fp_b2     = (const float*)d_in[5];
  const float* ln_g      = (const float*)d_in[6];
  const float* ln_b      = (const float*)d_in[7];
  const float* conv1_w   = (const float*)d_in[8];
  const float* conv1_b   = (const float*)d_in[9];
  const float* conv2_w   = (const float*)d_in[10];
  const float* conv2_b   = (const float*)d_in[11];
  const float* sp_w      = (const float*)d_in[12];
  const float* sp_b      = (const float*)d_in[13];
  const float* attn_wq   = (const float*)d_in[14];
  const float* attn_wk   = (const float*)d_in[15];
  const float* attn_wv   = (const float*)d_in[16];
  const float* attn_wo   = (const float*)d_in[17];

  float* out  = (float*)d_out;            // [acts 128*64][holo 128*32768]
  float* ws   = (float*)d_ws;
  float* acts_out = out;
  float* holo_out = out + BB * TT;

  float* Fr   = ws + OFF_FR;
  float* Fi   = ws + OFF_FI;
  float* Fn   = ws + OFF_FN;
  float* Yr   = ws + OFF_YR;
  float* Yi   = ws + OFF_YI;
  float* mag  = ws + OFF_MAG;
  float* nrm  = ws + OFF_NRM;
  float* rnrm = ws + OFF_RNRM;
  float* fm   = ws + OFF_FM;
  float* hid  = ws + OFF_HID;
  float* sims = ws + OFF_SIMS;
  float* actw = ws + OFF_ACTS;
  float* p1   = ws + OFF_P1;
  float* sft  = ws + OFF_S;
  float* z    = ws + OFF_Z;
  float* mu   = ws + OFF_MU;
  float* rstd = ws + OFF_RSTD;

  // frequency path
  make_twiddles<<<64, 256, 0, stream>>>(Fr, Fi, Fn);
  dft_stage1<<<NPLANE, 256, 0, stream>>>(image, Fr, Fi, Yr, Yi);
  dft_stage2<<<NPLANE, 256, 0, stream>>>(Yr, Yi, Fr, Fi, Fn, mag);
  mag_stats<<<BB, 256, 0, stream>>>(mag, nrm, fm);
  recept_norms<<<TT, 256, 0, stream>>>(receptors, rnrm);
  sims_gemm<<<1, 256, 0, stream>>>(mag, receptors, nrm, rnrm, sims);
  attention_acts<<<BB, 64, 0, stream>>>(sims, attn_wq, attn_wk, attn_wv, attn_wo,
                                        acts_out, actw);

  // spatial path
  conv1_pool<<<BB * 32, 256, 0, stream>>>(image, conv1_w, conv1_b, p1);
  conv2_gap<<<BB * 64, 256, 0, stream>>>(p1, conv2_w, conv2_b, sft);

  // frequency processor + layernorm
  fp1_gemm<<<8, 256, 0, stream>>>(fm, fp_w1, fp_b1, hid);
  fp2_gemm<<<512, 256, 0, stream>>>(hid, fp_w2, fp_b2, z);
  ln_stats<<<BB, 256, 0, stream>>>(z, mu, rstd);

  // holographic combine
  holo_gemm<<<512, 256, 0, stream>>>(z, mu, rstd, ln_g, ln_b,
                                     actw, receptors, sft, sp_w, sp_b, holo_out);
}